// GNNModel_74560632259283
// MI455X (gfx1250) — compile-verified
//
#include <hip/hip_runtime.h>
#include <hip/hip_bf16.h>
#include <math.h>

typedef __attribute__((ext_vector_type(2))) float v2f;
typedef __attribute__((ext_vector_type(8))) float v8f;

#define NNODES 100000
#define NEDGES 1600000
#define HDIM   64
#define DEDGE  16
#define NCLS   15
#define KFC    144   // 2*H + D_EDGE

// single-instruction ReLU: v_max_num_f32 already implements IEEE maximumNumber,
// so bypass LLVM's canonicalize+select expansion of fmaxf.
__device__ __forceinline__ float relu1(float x) {
    float r;
    asm("v_max_num_f32 %0, %1, 0" : "=v"(r) : "v"(x));
    return r;
}

// ---------------- degree / normalization ----------------

__global__ void k_deg_init(float* __restrict__ deg) {
    int i = blockIdx.x * blockDim.x + threadIdx.x;
    if (i < NNODES) deg[i] = 1.0f;              // self-loop contribution
}

__global__ void k_deg_acc(const int* __restrict__ dst, float* __restrict__ deg) {
    int e = blockIdx.x * blockDim.x + threadIdx.x;
    if (e < NEDGES) unsafeAtomicAdd(&deg[dst[e]], 1.0f);
}

__global__ void k_rsqrt_inplace(float* __restrict__ deg) {
    int i = blockIdx.x * blockDim.x + threadIdx.x;
    if (i < NNODES) deg[i] = rsqrtf(deg[i]);    // deg >= 1 always
}

// ---------------- node GEMM: T = (relu?)A @ W  (A:[N,64], W:[64,64]) ----------------
// One wave computes a 16x64 output tile with fp32 WMMA 16x16x4.
// RELU is a compile-time flag: no per-element select in the hot loop.

template <int RELU>
__global__ void k_gemm64(const float* __restrict__ A, const float* __restrict__ W,
                         float* __restrict__ T) {
    int wave = (int)((blockIdx.x * blockDim.x + threadIdx.x) >> 5);
    int lane = threadIdx.x & 31;
    int row0 = wave * 16;
    if (row0 >= NNODES) return;                 // whole-wave exit, EXEC stays full for WMMA
    int r = lane & 15;                          // A row / B,C column within tile
    int g = lane >> 4;                          // lane-half selects K sub-pair / M+8
    v8f acc0 = {}, acc1 = {}, acc2 = {}, acc3 = {};
    const float* arow = A + (size_t)(row0 + r) * HDIM;
#pragma unroll
    for (int kk = 0; kk < 16; ++kk) {
        int k0 = kk * 4 + g * 2;
        float a0 = arow[k0], a1 = arow[k0 + 1];
        if (RELU) { a0 = relu1(a0); a1 = relu1(a1); }
        v2f a; a.x = a0; a.y = a1;
        const float* w0 = W + (size_t)k0 * HDIM;
        v2f b;
        b.x = w0[r];           b.y = w0[HDIM + r];
        acc0 = __builtin_amdgcn_wmma_f32_16x16x4_f32(false, a, false, b, (short)0, acc0, false, false);
        b.x = w0[16 + r];      b.y = w0[HDIM + 16 + r];
        acc1 = __builtin_amdgcn_wmma_f32_16x16x4_f32(false, a, false, b, (short)0, acc1, false, false);
        b.x = w0[32 + r];      b.y = w0[HDIM + 32 + r];
        acc2 = __builtin_amdgcn_wmma_f32_16x16x4_f32(false, a, false, b, (short)0, acc2, false, false);
        b.x = w0[48 + r];      b.y = w0[HDIM + 48 + r];
        acc3 = __builtin_amdgcn_wmma_f32_16x16x4_f32(false, a, false, b, (short)0, acc3, false, false);
    }
    float* out = T + (size_t)(row0 + g * 8) * HDIM;
#pragma unroll
    for (int rr = 0; rr < 8; ++rr) {
        float* o = out + (size_t)rr * HDIM;
        o[r] = acc0[rr]; o[16 + r] = acc1[rr]; o[32 + r] = acc2[rr]; o[48 + r] = acc3[rr];
    }
}

// ---------------- aggregation init: G = (1/deg)*T + b  (self-loop + bias) ----------------
// float4-vectorized: one thread handles 4 contiguous features.

__global__ void k_initagg(const float* __restrict__ T, const float* __restrict__ dis,
                          const float* __restrict__ b, float* __restrict__ G) {
    int idx = blockIdx.x * blockDim.x + threadIdx.x;          // over NNODES*16
    if (idx >= NNODES * (HDIM / 4)) return;
    int i = idx >> 4;
    int k4 = (idx & 15) * 4;
    float dsc = dis[i]; dsc *= dsc;                            // 1/deg
    const float4 t  = *(const float4*)(T + (size_t)i * HDIM + k4);
    const float4 bb = *(const float4*)(b + k4);
    float4 o;
    o.x = dsc * t.x + bb.x;  o.y = dsc * t.y + bb.y;
    o.z = dsc * t.z + bb.z;  o.w = dsc * t.w + bb.w;
    *(float4*)(G + (size_t)i * HDIM + k4) = o;
}

// ---------------- edge scatter: G[dst] += d^-1/2[s]*d^-1/2[d] * T[src] ----------------
// One wave per edge; e forced uniform so index/norm loads become scalar (SMEM).

__global__ void k_scatter(const float* __restrict__ T, const int* __restrict__ src,
                          const int* __restrict__ dst, const float* __restrict__ dis,
                          float* __restrict__ G) {
    int idx = blockIdx.x * blockDim.x + threadIdx.x;
    int e = __builtin_amdgcn_readfirstlane(idx >> 5);          // wave-uniform edge id
    if (e >= NEDGES) return;
    int lane = threadIdx.x & 31;
    int s = src[e], d = dst[e];                                // scalar loads
    float nrm = dis[s] * dis[d];
    const float2 t = *(const float2*)(T + (size_t)s * HDIM + lane * 2);   // b64 load
    float* grow = G + (size_t)d * HDIM + lane * 2;
    unsafeAtomicAdd(grow,     nrm * t.x);
    unsafeAtomicAdd(grow + 1, nrm * t.y);
}

// ---------------- edge classifier + log_softmax ----------------
// One wave per 16-edge tile: feat = [relu(G[src]) | relu(G[dst]) | edge_attr] (K=144),
// logits = feat @ Wfc_padded + bfc, then row log_softmax over 15 valid cols.
// Wfc staged in LDS zero-padded to 16 columns -> unmasked conflict-free ds loads.

__global__ void k_edgefc(const float* __restrict__ G, const int* __restrict__ src,
                         const int* __restrict__ dst, const float* __restrict__ eattr,
                         const float* __restrict__ Wfc, const float* __restrict__ bfc,
                         float* __restrict__ out) {
    __shared__ float sW[KFC * 16];            // [k][c], c padded to 16 (col 15 = 0)
    __shared__ float sb[16];
    for (int i = threadIdx.x; i < KFC * 16; i += 256) {
        int k = i >> 4, c = i & 15;
        sW[i] = (c < NCLS) ? Wfc[(size_t)k * NCLS + c] : 0.f;
    }
    if (threadIdx.x < 16) sb[threadIdx.x] = (threadIdx.x < NCLS) ? bfc[threadIdx.x] : 0.f;
    __syncthreads();

    int wave = (int)((blockIdx.x * blockDim.x + threadIdx.x) >> 5);
    int lane = threadIdx.x & 31;
    int e0 = wave * 16;
    if (e0 >= NEDGES) return;                  // whole-wave exit (after barrier)
    int r = lane & 15;
    int g = lane >> 4;
    int e = e0 + r;                            // this lane's A-matrix row (edge)
    int s = src[e], d = dst[e];
    const float* hs = G + (size_t)s * HDIM;
    const float* hd = G + (size_t)d * HDIM;
    const float* ea = eattr + (size_t)e * DEDGE;
    v8f acc = {};
#pragma unroll
    for (int kk = 0; kk < 36; ++kk) {
        int k0 = kk * 4 + g * 2;               // feature segment fixed per kk
        float a0, a1;
        if (kk < 16)      { a0 = relu1(hs[k0]);      a1 = relu1(hs[k0 + 1]); }
        else if (kk < 32) { a0 = relu1(hd[k0 - 64]); a1 = relu1(hd[k0 - 63]); }
        else              { a0 = ea[k0 - 128];       a1 = ea[k0 - 127]; }
        v2f a; a.x = a0; a.y = a1;
        v2f b;
        b.x = sW[k0 * 16 + r];
        b.y = sW[k0 * 16 + 16 + r];
        acc = __builtin_amdgcn_wmma_f32_16x16x4_f32(false, a, false, b, (short)0, acc, false, false);
    }
    float bias = sb[r];
#pragma unroll
    for (int rr = 0; rr < 8; ++rr) {
        int erow = e0 + rr + g * 8;
        float v = acc[rr] + bias;
        float vm = (r < NCLS) ? v : -INFINITY;
#pragma unroll
        for (int off = 1; off < 16; off <<= 1) vm = fmaxf(vm, __shfl_xor(vm, off, 16));
        float ex = (r < NCLS) ? __expf(v - vm) : 0.f;
        float sum = ex;
#pragma unroll
        for (int off = 1; off < 16; off <<= 1) sum += __shfl_xor(sum, off, 16);
        if (r < NCLS) out[(size_t)erow * NCLS + r] = v - vm - __logf(sum);
    }
}

// ---------------- host launch ----------------

extern "C" void kernel_launch(void* const* d_in, const int* in_sizes, int n_in,
                              void* d_out, int out_size, void* d_ws, size_t ws_size,
                              hipStream_t stream) {
    const float* x     = (const float*)d_in[0];
    const int*   ei    = (const int*)  d_in[1];
    const float* eattr = (const float*)d_in[2];
    const float* W1 = (const float*)d_in[3];  const float* b1 = (const float*)d_in[4];
    const float* W2 = (const float*)d_in[5];  const float* b2 = (const float*)d_in[6];
    const float* W3 = (const float*)d_in[7];  const float* b3 = (const float*)d_in[8];
    const float* Wfc = (const float*)d_in[9]; const float* bfc = (const float*)d_in[10];
    float* out = (float*)d_out;
    const int* src = ei;
    const int* dst = ei + NEDGES;

    char* ws = (char*)d_ws;
    float* dis = (float*)ws;                                           // N floats
    float* T   = (float*)(ws + (1u << 20));                            // N*H floats
    float* G   = (float*)(ws + (1u << 20) + (size_t)NNODES * HDIM * sizeof(float));

    // normalization: deg -> d^-1/2 (in place)
    k_deg_init<<<(NNODES + 255) / 256, 256, 0, stream>>>(dis);
    k_deg_acc <<<(NEDGES + 255) / 256, 256, 0, stream>>>(dst, dis);
    k_rsqrt_inplace<<<(NNODES + 255) / 256, 256, 0, stream>>>(dis);

    const int mtiles      = NNODES / 16;                    // 6250
    const int gemm_blocks = (mtiles + 7) / 8;               // 8 waves / block
    const int init_blocks = (NNODES * (HDIM / 4)) / 256;    // 6250
    const int scat_blocks = (NEDGES * 32) / 256;            // 200000 (1 wave / edge)

    // layer 1
    k_gemm64<0><<<gemm_blocks, 256, 0, stream>>>(x, W1, T);
    k_initagg  <<<init_blocks, 256, 0, stream>>>(T, dis, b1, G);
    k_scatter  <<<scat_blocks, 256, 0, stream>>>(T, src, dst, dis, G);
    // layer 2 (relu applied on GEMM input load)
    k_gemm64<1><<<gemm_blocks, 256, 0, stream>>>(G, W2, T);
    k_initagg  <<<init_blocks, 256, 0, stream>>>(T, dis, b2, G);
    k_scatter  <<<scat_blocks, 256, 0, stream>>>(T, src, dst, dis, G);
    // layer 3
    k_gemm64<1><<<gemm_blocks, 256, 0, stream>>>(G, W3, T);
    k_initagg  <<<init_blocks, 256, 0, stream>>>(T, dis, b3, G);
    k_scatter  <<<scat_blocks, 256, 0, stream>>>(T, src, dst, dis, G);

    // edge classifier + log_softmax (relu of layer-3 applied on gather)
    const int fc_blocks = (NEDGES / 16) / 8;                // 12500
    k_edgefc<<<fc_blocks, 256, 0, stream>>>(G, src, dst, eattr, Wfc, bfc, out);
}